// ViT_86423331930597
// MI455X (gfx1250) — compile-verified
//
#include <hip/hip_runtime.h>
#include <hip/hip_bf16.h>
#include <cstdint>

// ---------------------------------------------------------------------------
// CDNA5 fp32 WMMA (V_WMMA_F32_16X16X4_F32, wave32)
// A lane layout (16x4):  a[j] = A[l&15][((l>>4)<<1)+j]
// B lane layout (4x16):  b[j] = B[((l>>4)<<1)+j][l&15]
// C/D  (16x16, 8 VGPR):  c[i] = C[(l>>4)*8 + i][l&15]
// ---------------------------------------------------------------------------
typedef __attribute__((ext_vector_type(2))) float v2f;
typedef __attribute__((ext_vector_type(8))) float v8f;

#define KC 64        // K chunk staged in LDS per stage
#define PAD_A 68     // row stride (floats), banks 4m/4m+2 -> conflict-free
#define PAD_NK 68    // B stored n-major [64 n][68 k]
#define PAD_KN 72    // B stored k-major [64 k][72 n]; 2*72%64=16 -> conflict-free

__device__ __forceinline__ v8f wmma_f32_16x16x4(v2f a, v2f b, v8f c) {
    return __builtin_amdgcn_wmma_f32_16x16x4_f32(false, a, false, b, (short)0, c,
                                                 false, false);
}

// ---- CDNA5 async global->LDS DMA (ASYNCcnt tracked, bypasses VGPRs) -------
__device__ __forceinline__ void async_b128(void* lds, const float* g) {
    // low 32 bits of a generic pointer to __shared__ == LDS byte offset
    asm volatile("global_load_async_to_lds_b128 %0, %1, off"
                 :: "v"((uint32_t)(uintptr_t)lds), "v"(g)
                 : "memory");
}
template <int N>
__device__ __forceinline__ void wait_async_le() {
    asm volatile("s_wait_asynccnt %0" :: "i"(N) : "memory");
}

__device__ __forceinline__ float gelu_exact(float x) {
    return 0.5f * x * (1.0f + erff(x * 0.70710678118654752f));
}

// ---------------------------------------------------------------------------
// Generic tiled WMMA GEMM: C[M,N] = epi(A[M,K] @ B)  (K multiple of 64)
//   B_NK      : B stored [N][K] (wq/wk/wv) else [K][N] row-major
//   BIAS/RES/GELU : fused epilogue (+bias, gelu, +residual, in that order)
//   STORE_BHS : scatter C[m=b*257+s][n=h] to out[((b*N+n)*257)+s] (head-major)
// Block: 128 threads = 4 waves; tile 16(M) x 64(N); wave w owns n-subtile w.
// Double-buffered async-to-LDS staging: 10 async instrs / wave / chunk.
// ---------------------------------------------------------------------------
template <bool B_NK, bool BIAS, bool RES, bool GELU, bool STORE_BHS>
__global__ __launch_bounds__(128) void gemm16_wmma(
    const float* __restrict__ A, const float* __restrict__ Bm,
    const float* __restrict__ bias, const float* __restrict__ resid,
    float* __restrict__ C, int M, int N, int K) {
    constexpr int BP = B_NK ? PAD_NK : PAD_KN;
    __shared__ float As[2][16][PAD_A];
    __shared__ float Bs[2][64][BP];

    const int tid = threadIdx.x;
    const int w = tid >> 5;
    const int l = tid & 31;
    const int m0 = blockIdx.x * 16;
    const int n0 = blockIdx.y * 64;
    const int mrow = l & 15;
    const int kh = (l >> 4) << 1;

    auto stage = [&](int buf, int kc) {
        // A tile 16 x 64: rows beyond M clamped (garbage rows are discarded)
        for (int idx = tid; idx < 16 * 16; idx += 128) {
            const int r = idx >> 4, c4 = idx & 15;
            const int rr = (m0 + r < M) ? (m0 + r) : (M - 1);
            async_b128(&As[buf][r][c4 * 4], A + (size_t)rr * K + kc + c4 * 4);
        }
        if (B_NK) {  // contiguous along k -> n-major LDS tile
            for (int idx = tid; idx < 64 * 16; idx += 128) {
                const int j = idx >> 4, k4 = idx & 15;
                async_b128(&Bs[buf][j][k4 * 4], Bm + (size_t)(n0 + j) * K + kc + k4 * 4);
            }
        } else {     // contiguous along n -> k-major LDS tile
            for (int idx = tid; idx < 64 * 16; idx += 128) {
                const int kk = idx >> 4, j4 = idx & 15;
                async_b128(&Bs[buf][kk][j4 * 4], Bm + (size_t)(kc + kk) * N + n0 + j4 * 4);
            }
        }
    };

    v8f c = {};
    const int nch = K / KC;
    stage(0, 0);
    for (int ic = 0; ic < nch; ++ic) {
        const int buf = ic & 1;
        if (ic + 1 < nch) {
            stage(buf ^ 1, (ic + 1) * KC);
            wait_async_le<10>();   // current chunk's 10 DMAs complete
        } else {
            wait_async_le<0>();
        }
        __syncthreads();
#pragma unroll
        for (int kk = 0; kk < KC; kk += 4) {
            v2f a = {As[buf][mrow][kk + kh], As[buf][mrow][kk + kh + 1]};
            v2f b;
            if (B_NK)
                b = (v2f){Bs[buf][w * 16 + mrow][kk + kh], Bs[buf][w * 16 + mrow][kk + kh + 1]};
            else
                b = (v2f){Bs[buf][kk + kh][w * 16 + mrow], Bs[buf][kk + kh + 1][w * 16 + mrow]};
            c = wmma_f32_16x16x4(a, b, c);
        }
        __syncthreads();  // all waves done with buf before it is re-staged
    }

#pragma unroll
    for (int i = 0; i < 8; ++i) {
        const int m = m0 + ((l >> 4) << 3) + i;
        const int n = n0 + w * 16 + (l & 15);
        if (m < M) {
            float v = c[i];
            if (BIAS) v += bias[n];
            if (GELU) v = gelu_exact(v);
            if (RES) v += resid[(size_t)m * N + n];
            if (STORE_BHS) {
                const int b = m / 257, s = m % 257;
                C[((size_t)(b * N + n)) * 257 + s] = v;
            } else {
                C[(size_t)m * N + n] = v;
            }
        }
    }
}

// ---------------------------------------------------------------------------
// Patch embed: t[b, h+1, n] = sum_i p[b,n,i]*patch_w[h,n,i] + patch_b[h,n]
//                             + pos_emb[h+1, n]
// 256 independent 8x256x256 GEMMs (one per n). grid=(256 n, 4 h-groups),
// block=128 (4 waves, each a 16-wide h tile). M=8 padded to 16 (zero rows).
// Double-buffered async staging: 9 async instrs / wave / chunk.
// ---------------------------------------------------------------------------
__global__ __launch_bounds__(128) void patch_embed_wmma(
    const float* __restrict__ x, const float* __restrict__ pw,
    const float* __restrict__ pb, const float* __restrict__ pos,
    float* __restrict__ t) {
    __shared__ float As[2][16][PAD_A];
    __shared__ float Bs[2][64][PAD_NK];   // n-major: Bs[buf][h-j][k]

    const int n = blockIdx.x;            // patch index == output channel d
    const int h0 = blockIdx.y * 64;
    const int tid = threadIdx.x;
    const int w = tid >> 5;
    const int l = tid & 31;
    const int py = n >> 4, px = n & 15;
    const int mrow = l & 15;
    const int kh = (l >> 4) << 1;

    // zero rows 8..15 of both buffers once (DMA only ever writes rows 0..7)
    for (int idx = tid; idx < 2 * 8 * PAD_A; idx += 128) {
        const int bb = idx / (8 * PAD_A);
        const int rem = idx % (8 * PAD_A);
        As[bb][8 + rem / PAD_A][rem % PAD_A] = 0.f;
    }

    auto stage = [&](int buf, int kc) {
        // A: p[b][n][k], k = cy*16+cx gathered from image (1 DMA / wave)
        for (int idx = tid; idx < 8 * 16; idx += 128) {
            const int b = idx >> 4, c4 = idx & 15;
            const int k = kc + c4 * 4;
            const int cy = k >> 4, cx = k & 15;
            async_b128(&As[buf][b][c4 * 4],
                       x + ((size_t)(b * 256 + py * 16 + cy)) * 256 + px * 16 + cx);
        }
        // B: patch_w[h0+j][n][kc..] contiguous along k (8 DMAs / wave)
        for (int idx = tid; idx < 64 * 16; idx += 128) {
            const int j = idx >> 4, k4 = idx & 15;
            async_b128(&Bs[buf][j][k4 * 4],
                       pw + ((size_t)(h0 + j)) * 65536 + (size_t)n * 256 + kc + k4 * 4);
        }
    };

    v8f c = {};
    stage(0, 0);
    for (int ic = 0; ic < 4; ++ic) {
        const int buf = ic & 1;
        if (ic + 1 < 4) {
            stage(buf ^ 1, (ic + 1) * KC);
            wait_async_le<9>();
        } else {
            wait_async_le<0>();
        }
        __syncthreads();
#pragma unroll
        for (int kk = 0; kk < KC; kk += 4) {
            v2f a = {As[buf][mrow][kk + kh], As[buf][mrow][kk + kh + 1]};
            v2f b = {Bs[buf][w * 16 + mrow][kk + kh], Bs[buf][w * 16 + mrow][kk + kh + 1]};
            c = wmma_f32_16x16x4(a, b, c);
        }
        __syncthreads();
    }

    if (l < 16) {  // lanes 0..15 carry rows b = i (i < 8 valid)
        const int hh = h0 + w * 16 + l;
#pragma unroll
        for (int i = 0; i < 8; ++i) {
            float v = c[i] + pb[hh * 256 + n] + pos[(hh + 1) * 256 + n];
            t[((size_t)(i * 257 + hh + 1)) * 256 + n] = v;
        }
    }
}

// ---------------------------------------------------------------------------
// class-token row: t[b][0][d] = class_token[d] + pos_emb[0][d]
// ---------------------------------------------------------------------------
__global__ void cls_init(const float* __restrict__ ct,
                         const float* __restrict__ pos, float* __restrict__ t) {
    const int i = blockIdx.x * blockDim.x + threadIdx.x;  // 0..2047
    const int b = i >> 8, d = i & 255;
    t[((size_t)b * 257) * 256 + d] = ct[d] + pos[d];
}

// ---------------------------------------------------------------------------
// LayerNorm over D=256, one block per row (2056 rows)
// ---------------------------------------------------------------------------
__global__ __launch_bounds__(256) void layernorm_k(
    const float* __restrict__ in, const float* __restrict__ g,
    const float* __restrict__ bta, float* __restrict__ out) {
    __shared__ float red[256];
    const int row = blockIdx.x, tid = threadIdx.x;
    const float v = in[(size_t)row * 256 + tid];
    red[tid] = v;
    __syncthreads();
    for (int off = 128; off > 0; off >>= 1) {
        if (tid < off) red[tid] += red[tid + off];
        __syncthreads();
    }
    const float mu = red[0] * (1.0f / 256.0f);
    __syncthreads();
    const float d = v - mu;
    red[tid] = d * d;
    __syncthreads();
    for (int off = 128; off > 0; off >>= 1) {
        if (tid < off) red[tid] += red[tid + off];
        __syncthreads();
    }
    const float var = red[0] * (1.0f / 256.0f);
    out[(size_t)row * 256 + tid] = d * rsqrtf(var + 1e-5f) * g[tid] + bta[tid];
}

// ---------------------------------------------------------------------------
// Attention, DH=1 (scores are rank-1 outer products): per (b,h) block,
// streaming stable softmax. Q/K/V head-major [B][H][S] -> contiguous.
// ---------------------------------------------------------------------------
__global__ __launch_bounds__(256) void attention_k(
    const float* __restrict__ Qt, const float* __restrict__ Kt,
    const float* __restrict__ Vt, float* __restrict__ Ot) {
    __shared__ float ks[257];
    __shared__ float vs[257];
    __shared__ float red[256];
    const size_t base = (size_t)blockIdx.x * 257;
    const int tid = threadIdx.x;

    float lmax = -1e30f, lmin = 1e30f;
    for (int s = tid; s < 257; s += 256) {
        const float kv = Kt[base + s];
        ks[s] = kv;
        vs[s] = Vt[base + s];
        lmax = fmaxf(lmax, kv);
        lmin = fminf(lmin, kv);
    }
    red[tid] = lmax;
    __syncthreads();
    for (int off = 128; off > 0; off >>= 1) {
        if (tid < off) red[tid] = fmaxf(red[tid], red[tid + off]);
        __syncthreads();
    }
    const float kmax = red[0];
    __syncthreads();
    red[tid] = lmin;
    __syncthreads();
    for (int off = 128; off > 0; off >>= 1) {
        if (tid < off) red[tid] = fminf(red[tid], red[tid + off]);
        __syncthreads();
    }
    const float kmin = red[0];
    __syncthreads();

    for (int s = tid; s < 257; s += 256) {
        const float q = Qt[base + s];
        const float mx = (q >= 0.f) ? q * kmax : q * kmin;
        float num = 0.f, den = 0.f;
        for (int u = 0; u < 257; ++u) {
            const float e = __expf(q * ks[u] - mx);
            num = fmaf(e, vs[u], num);
            den += e;
        }
        Ot[base + s] = num / den;
    }
}

// ---------------------------------------------------------------------------
// Transpose o: [B][H][S] -> [B*S][H] via 16x16 LDS tiles
// ---------------------------------------------------------------------------
__global__ __launch_bounds__(256) void transpose_o_k(const float* __restrict__ Ot,
                                                     float* __restrict__ O) {
    __shared__ float tile[16][17];
    const int b = blockIdx.z;
    const int s0 = blockIdx.x * 16;
    const int h0 = blockIdx.y * 16;
    const int tx = threadIdx.x & 15, ty = threadIdx.x >> 4;
    if (s0 + tx < 257)
        tile[ty][tx] = Ot[((size_t)(b * 256 + h0 + ty)) * 257 + s0 + tx];
    __syncthreads();
    if (s0 + ty < 257)
        O[((size_t)(b * 257 + s0 + ty)) * 256 + h0 + tx] = tile[tx][ty];
}

// ---------------------------------------------------------------------------
// Classifier head + final softmax
// ---------------------------------------------------------------------------
__global__ __launch_bounds__(256) void head_k(const float* __restrict__ t,
                                              const float* __restrict__ hw,
                                              const float* __restrict__ hb,
                                              float* __restrict__ logits) {
    const int b = blockIdx.x;
    const int c = blockIdx.y * 256 + threadIdx.x;
    if (c >= 1000) return;
    const float* row = t + (size_t)b * 257 * 256;
    float acc = hb[c];
#pragma unroll 4
    for (int d = 0; d < 256; ++d) acc = fmaf(row[d], hw[d * 1000 + c], acc);
    logits[b * 1000 + c] = acc;
}

__global__ __launch_bounds__(256) void softmax1000_k(const float* __restrict__ L,
                                                     float* __restrict__ out) {
    __shared__ float red[256];
    const int b = blockIdx.x, tid = threadIdx.x;
    const float* row = L + b * 1000;
    float lm = -1e30f;
    for (int c = tid; c < 1000; c += 256) lm = fmaxf(lm, row[c]);
    red[tid] = lm;
    __syncthreads();
    for (int off = 128; off > 0; off >>= 1) {
        if (tid < off) red[tid] = fmaxf(red[tid], red[tid + off]);
        __syncthreads();
    }
    const float mx = red[0];
    __syncthreads();
    float ls = 0.f;
    for (int c = tid; c < 1000; c += 256) ls += __expf(row[c] - mx);
    red[tid] = ls;
    __syncthreads();
    for (int off = 128; off > 0; off >>= 1) {
        if (tid < off) red[tid] += red[tid + off];
        __syncthreads();
    }
    const float inv = 1.0f / red[0];
    for (int c = tid; c < 1000; c += 256) out[b * 1000 + c] = __expf(row[c] - mx) * inv;
}

// ---------------------------------------------------------------------------
extern "C" void kernel_launch(void* const* d_in, const int* in_sizes, int n_in,
                              void* d_out, int out_size, void* d_ws, size_t ws_size,
                              hipStream_t stream) {
    const float* x           = (const float*)d_in[0];
    const float* class_token = (const float*)d_in[1];
    const float* pos_emb     = (const float*)d_in[2];
    const float* patch_w     = (const float*)d_in[3];
    const float* patch_b     = (const float*)d_in[4];
    const float* wq          = (const float*)d_in[5];
    const float* wk          = (const float*)d_in[6];
    const float* wv          = (const float*)d_in[7];
    const float* proj_w      = (const float*)d_in[8];
    const float* proj_b      = (const float*)d_in[9];
    const float* mlp_w1      = (const float*)d_in[10];
    const float* mlp_b1      = (const float*)d_in[11];
    const float* mlp_w2      = (const float*)d_in[12];
    const float* mlp_b2      = (const float*)d_in[13];
    const float* ln1_g       = (const float*)d_in[14];
    const float* ln1_b       = (const float*)d_in[15];
    const float* ln2_g       = (const float*)d_in[16];
    const float* ln2_b       = (const float*)d_in[17];
    const float* head_w      = (const float*)d_in[18];
    const float* head_b      = (const float*)d_in[19];

    const size_t R = (size_t)8 * 257 * 256;  // 526336 floats
    float* ws     = (float*)d_ws;
    float* t      = ws;               // [B*S][D] residual stream
    float* xn     = t + R;            // LN output (reused for ln1 & ln2)
    float* Qt     = xn + R;           // [B][H][S] head-major
    float* Kt     = Qt + R;
    float* Vt     = Kt + R;
    float* Ot     = Vt + R;           // attention out, head-major
    float* Oc     = Ot + R;           // attention out, [B*S][H]
    float* hbuf   = Oc + R;           // [B*S][MLP]
    float* logits = hbuf + (size_t)8 * 257 * 1024;

    const int M = 8 * 257;  // 2056
    const dim3 g256(129, 4), gmlp(129, 16);

    // 1. residual stream init
    cls_init<<<8, 256, 0, stream>>>(class_token, pos_emb, t);
    patch_embed_wmma<<<dim3(256, 4), 128, 0, stream>>>(x, patch_w, patch_b, pos_emb, t);

    // 2. LN1 (shared by Q,K,V)
    layernorm_k<<<M, 256, 0, stream>>>(t, ln1_g, ln1_b, xn);

    // 3. QKV GEMMs, weights stored [H][D] (NK), outputs head-major [B][H][S]
    gemm16_wmma<true, false, false, false, true>
        <<<g256, 128, 0, stream>>>(xn, wq, nullptr, nullptr, Qt, M, 256, 256);
    gemm16_wmma<true, false, false, false, true>
        <<<g256, 128, 0, stream>>>(xn, wk, nullptr, nullptr, Kt, M, 256, 256);
    gemm16_wmma<true, false, false, false, true>
        <<<g256, 128, 0, stream>>>(xn, wv, nullptr, nullptr, Vt, M, 256, 256);

    // 4. streaming rank-1 attention, one block per (b,h)
    attention_k<<<2048, 256, 0, stream>>>(Qt, Kt, Vt, Ot);
    transpose_o_k<<<dim3(17, 16, 8), 256, 0, stream>>>(Ot, Oc);

    // 5. out-proj with fused bias + residual (in-place on t)
    gemm16_wmma<false, true, true, false, false>
        <<<g256, 128, 0, stream>>>(Oc, proj_w, proj_b, t, t, M, 256, 256);

    // 6. LN2 + MLP (fused exact-GELU), residual back into t
    layernorm_k<<<M, 256, 0, stream>>>(t, ln2_g, ln2_b, xn);
    gemm16_wmma<false, true, false, true, false>
        <<<gmlp, 128, 0, stream>>>(xn, mlp_w1, mlp_b1, nullptr, hbuf, M, 1024, 256);
    gemm16_wmma<false, true, true, false, false>
        <<<g256, 128, 0, stream>>>(hbuf, mlp_w2, mlp_b2, t, t, M, 256, 1024);

    // 7. classifier head + softmax
    head_k<<<dim3(8, 4), 256, 0, stream>>>(t, head_w, head_b, logits);
    softmax1000_k<<<8, 256, 0, stream>>>(logits, (float*)d_out);
}